// GCNLayer_42975442764291
// MI455X (gfx1250) — compile-verified
//
#include <hip/hip_runtime.h>
#include <hip/hip_bf16.h>

#define N_NODES 50000
#define N_EDGES 1250000
#define D 64   // D_IN == D_OUT == 64

typedef __attribute__((ext_vector_type(2))) float v2f;
typedef __attribute__((ext_vector_type(8))) float v8f;

// ---------------------------------------------------------------------------
// Kernel 1: zero the h accumulator (float4 grid-stride).
// ---------------------------------------------------------------------------
__global__ void gcn_zero_kernel(float4* __restrict__ h4, int n4) {
    int i = blockIdx.x * blockDim.x + threadIdx.x;
    int stride = gridDim.x * blockDim.x;
    float4 z = {0.f, 0.f, 0.f, 0.f};
    for (; i < n4; i += stride) h4[i] = z;
}

// ---------------------------------------------------------------------------
// Kernel 2: edge-parallel gather + scatter-add.
// 16 threads per edge; each thread moves one float4 chunk of the 64-float row.
// feature (12.8MB) and h (12.8MB) are L2-resident on MI455X (192MB L2), so
// this phase is L2-atomic-throughput bound; we minimize instruction count with
// b128 gathers and issue 4 non-returning f32 atomics per thread.
// ---------------------------------------------------------------------------
__global__ void gcn_scatter_kernel(const float* __restrict__ feature,
                                   const int* __restrict__ src,
                                   const int* __restrict__ dst,
                                   float* __restrict__ h) {
    long long gid = (long long)blockIdx.x * blockDim.x + threadIdx.x;
    int e = (int)(gid >> 4);           // edge index
    int c = (int)(gid & 15);           // chunk index within the 64-float row
    if (e >= N_EDGES) return;

    int s = src[e];
    int d = dst[e];

    const float4 v = *(const float4*)(feature + (long long)s * D + c * 4);
    float* out = h + (long long)d * D + c * 4;

    // Relaxed, device-scope fp32 atomic adds -> global_atomic_add_f32 (no rtn).
    __hip_atomic_fetch_add(out + 0, v.x, __ATOMIC_RELAXED, __HIP_MEMORY_SCOPE_AGENT);
    __hip_atomic_fetch_add(out + 1, v.y, __ATOMIC_RELAXED, __HIP_MEMORY_SCOPE_AGENT);
    __hip_atomic_fetch_add(out + 2, v.z, __ATOMIC_RELAXED, __HIP_MEMORY_SCOPE_AGENT);
    __hip_atomic_fetch_add(out + 3, v.w, __ATOMIC_RELAXED, __HIP_MEMORY_SCOPE_AGENT);
}

// ---------------------------------------------------------------------------
// Kernel 3: out = h @ W.T + b using V_WMMA_F32_16X16X4_F32 (full f32 math,
// matching the f32 reference). One wave computes a 16x16 output tile.
// 4 waves per block cover the 4 N-tiles of the 64-wide output; 3125 blocks
// cover the 50000 rows (50000 = 3125 * 16 exactly).
//
// VGPR layouts (CDNA5 ISA 7.12.2):
//   A 16x4 f32 (2 VGPRs): lanes 0-15 -> M=lane, K={0,1}; lanes 16-31 -> K={2,3}
//   B 4x16 f32 (2 VGPRs): lanes 0-15 -> N=lane, K={0,1}; lanes 16-31 -> K={2,3}
//   C/D 16x16 f32 (8 VGPRs): element [M = v + 8*(lane>=16)][N = lane&15]
// B[k][n] = (W^T)[k][n] = W[n][k].
// ---------------------------------------------------------------------------
__global__ void gcn_gemm_wmma_kernel(const float* __restrict__ h,
                                     const float* __restrict__ W,
                                     const float* __restrict__ b,
                                     float* __restrict__ out) {
    const int lane = threadIdx.x & 31;
    const int wave = threadIdx.x >> 5;      // 0..3 -> N tile
    const int half = lane >> 4;             // 0: lanes 0-15, 1: lanes 16-31
    const int l15  = lane & 15;

    const int row0 = blockIdx.x * 16;       // M tile base (node rows)
    const int n0   = wave * 16;             // N tile base (output cols)

    // Per-lane base pointers for A and B fragments (k offset 2*half per ISA).
    const float* Arow = h + (long long)(row0 + l15) * D + 2 * half;
    const float* Brow = W + (long long)(n0 + l15) * D + 2 * half;  // W[n][k]

    v8f c = {};
#pragma unroll
    for (int k0 = 0; k0 < D; k0 += 4) {
        v2f a  = *(const v2f*)(Arow + k0);  // A[M][k0+2h], A[M][k0+2h+1]
        v2f bf = *(const v2f*)(Brow + k0);  // B[k0+2h][N], B[k0+2h+1][N]
        c = __builtin_amdgcn_wmma_f32_16x16x4_f32(
                /*neg_a=*/false, a, /*neg_b=*/false, bf,
                /*c_mod=*/(short)0, c, /*reuse_a=*/false, /*reuse_b=*/false);
    }

    const float bias = b[n0 + l15];
#pragma unroll
    for (int v = 0; v < 8; ++v) {
        const int m = v + 8 * half;
        out[(long long)(row0 + m) * D + n0 + l15] = c[v] + bias;
    }
}

// ---------------------------------------------------------------------------
extern "C" void kernel_launch(void* const* d_in, const int* in_sizes, int n_in,
                              void* d_out, int out_size, void* d_ws, size_t ws_size,
                              hipStream_t stream) {
    const float* feature = (const float*)d_in[0];   // [N_NODES, D]
    const int*   src     = (const int*)d_in[1];     // [N_EDGES]
    const int*   dst     = (const int*)d_in[2];     // [N_EDGES]
    const float* W       = (const float*)d_in[3];   // [D, D]
    const float* b       = (const float*)d_in[4];   // [D]
    float*       out     = (float*)d_out;           // [N_NODES, D]

    float* h = (float*)d_ws;                        // accumulator, 12.8 MB

    // 1) zero h
    {
        int n4 = N_NODES * D / 4;                   // 800000 float4
        int threads = 256;
        int blocks = (n4 + threads - 1) / threads;  // 3125
        gcn_zero_kernel<<<blocks, threads, 0, stream>>>((float4*)h, n4);
    }

    // 2) gather + scatter-add over edges (16 threads / edge)
    {
        long long total = (long long)N_EDGES * 16;  // 20M threads
        int threads = 256;
        long long blocks = (total + threads - 1) / threads; // 78125
        gcn_scatter_kernel<<<(int)blocks, threads, 0, stream>>>(feature, src, dst, h);
    }

    // 3) WMMA f32 GEMM: out = h @ W.T + b
    {
        int blocks = N_NODES / 16;                  // 3125 (exact)
        gcn_gemm_wmma_kernel<<<blocks, 128, 0, stream>>>(h, W, b, out);
    }
}